// DynamicVoxelEncoder_25950192402987
// MI455X (gfx1250) — compile-verified
//
#include <hip/hip_runtime.h>
#include <cstdint>

// ---------------------------------------------------------------------------
// DynamicVoxelEncoder for MI455X (gfx1250)
//
// Memory-bound scatter-mean. No matrix structure -> WMMA not applicable.
// CDNA5 path used: async global->LDS copy engine (global_load_async_to_lds_b32,
// tracked by ASYNCcnt / s_wait_asynccnt) for double-buffered point streaming.
//
// d_out layout (flat, reference return order):
//   [0                 .. 46,656,000)  voxels  [B=4][NV=2,332,800][F=5]  f32
//   [46,656,000        .. 55,987,200)  counts  [B][NV]                   f32
//   [55,987,200        .. 55,987,203)  shape   (540, 540, 8)
// ---------------------------------------------------------------------------

namespace {
constexpr int   kNX = 540, kNY = 540, kNZ = 8;
constexpr int   kNV = kNX * kNY * kNZ;          // 2,332,800
constexpr int   kB  = 4;
constexpr int   kN  = 300000;
constexpr int   kF  = 5;
constexpr long long kVoxElems = (long long)kB * kNV * kF;  // 46,656,000
constexpr long long kCntElems = (long long)kB * kNV;       //  9,331,200
constexpr int   kTile = 256;                    // points per tile (== blockDim)
}

// --- CDNA5 async global->LDS copy (inline asm: portable across toolchains) ---
__device__ __forceinline__ void async_g2l_b32(const float* sbase, unsigned goff,
                                              unsigned ldsoff) {
  // GVS form: copies 4 bytes from global memory at saddr plus the offset in vaddr to the LDS address in vdst; tracked by ASYNCcnt.
  asm volatile("global_load_async_to_lds_b32 %0, %1, %2 offset:0"
               :: "v"(ldsoff), "v"(goff), "s"(sbase)
               : "memory");
}
#define WAIT_ASYNCCNT_0() asm volatile("s_wait_asynccnt 0x0" ::: "memory")
#define WAIT_ASYNCCNT_5() asm volatile("s_wait_asynccnt 0x5" ::: "memory")

// ---------------------------------------------------------------------------
__global__ void voxel_zero_kernel(float4* __restrict__ out4, long long n4,
                                  float* __restrict__ out, long long shape_off) {
  long long i = (long long)blockIdx.x * blockDim.x + threadIdx.x;
  const long long stride = (long long)gridDim.x * blockDim.x;
  const float4 z = make_float4(0.f, 0.f, 0.f, 0.f);
  for (; i < n4; i += stride) out4[i] = z;
  if (blockIdx.x == 0 && threadIdx.x == 0) {
    out[shape_off + 0] = 540.0f;
    out[shape_off + 1] = 540.0f;
    out[shape_off + 2] = 8.0f;
  }
}

// ---------------------------------------------------------------------------
__global__ void voxel_scatter_kernel(const float* __restrict__ pts,
                                     float* __restrict__ sums,
                                     float* __restrict__ cnts,
                                     int total) {
  __shared__ float tile[2][kTile * kF];   // 10,240 B of the 320 KB/WGP LDS
  const int tid = threadIdx.x;
  const int tilesTotal = (total + kTile - 1) / kTile;

  // Low 32 bits of a generic LDS address == byte offset into LDS (flat rules).
  const unsigned ldsBase = (unsigned)(uintptr_t)&tile[0][0];

  // Issue 5 async b32 copies for this lane's point of tile t into buffer b.
  // Lane index is clamped (not masked) so every wave's stage issues exactly
  // 5 async instructions -> s_wait_asynccnt arithmetic below is exact.
  auto stage = [&](int t, int b) {
    int idx = t * kTile + tid;
    if (idx > total - 1) idx = total - 1;
    const unsigned goff = (unsigned)idx * 20u;                     // 5*4 bytes
    const unsigned loff = ldsBase + (unsigned)b * (kTile * kF * 4)
                                  + (unsigned)tid * 20u;
#pragma unroll
    for (int f = 0; f < kF; ++f)
      async_g2l_b32(pts, goff + 4u * f, loff + 4u * f);
  };

  int t = blockIdx.x;
  if (t < tilesTotal) stage(t, 0);
  int buf = 0;

  for (; t < tilesTotal; t += gridDim.x) {
    const int tn = t + gridDim.x;
    if (tn < tilesTotal) {
      stage(tn, buf ^ 1);      // prefetch next tile (ASYNCcnt: 5 old + 5 new)
      WAIT_ASYNCCNT_5();       // async loads complete in order: current 5 done
    } else {
      WAIT_ASYNCCNT_0();       // drain
    }

    const int idx = t * kTile + tid;
    if (idx < total) {
      const float* p = &tile[buf][tid * kF];
      const float x = p[0], y = p[1], z = p[2], f0 = p[3], f1 = p[4];

      const bool keep = (x >= -54.0f) && (x <= 54.0f) &&
                        (y >= -54.0f) && (y <= 54.0f) &&
                        (z >= -5.0f)  && (z <= 3.0f);
      if (keep) {
        // Exact IEEE divide + floor to match the JAX reference bit-for-bit.
        int cx = (int)floorf((x + 54.0f) / 0.2f);
        int cy = (int)floorf((y + 54.0f) / 0.2f);
        int cz = (int)floorf((z + 5.0f)  / 1.0f);
        cx = min(max(cx, 0), kNX - 1);
        cy = min(max(cy, 0), kNY - 1);
        cz = min(max(cz, 0), kNZ - 1);
        const int lin = (cz * kNY + cy) * kNX + cx;
        const int b   = idx / kN;
        const long long vi = (long long)b * kNV + lin;

        float* s = sums + vi * kF;
        atomicAdd(s + 0, x);
        atomicAdd(s + 1, y);
        atomicAdd(s + 2, z);
        atomicAdd(s + 3, f0);
        atomicAdd(s + 4, f1);
        atomicAdd(cnts + vi, 1.0f);
      }
    }
    buf ^= 1;
  }
}

// ---------------------------------------------------------------------------
__global__ void voxel_finalize_kernel(float* __restrict__ sums,
                                      const float* __restrict__ cnts,
                                      long long nvoxTotal) {
  const long long v = (long long)blockIdx.x * blockDim.x + threadIdx.x;
  if (v >= nvoxTotal) return;
  const float c = cnts[v];
  const float inv = 1.0f / fmaxf(c, 1.0f);
  float* s = sums + v * kF;
#pragma unroll
  for (int f = 0; f < kF; ++f) s[f] *= inv;
}

// ---------------------------------------------------------------------------
extern "C" void kernel_launch(void* const* d_in, const int* in_sizes, int n_in,
                              void* d_out, int out_size, void* d_ws, size_t ws_size,
                              hipStream_t stream) {
  (void)in_sizes; (void)n_in; (void)out_size; (void)d_ws; (void)ws_size;

  const float* pts = (const float*)d_in[0];            // [4,300000,5] f32
  float* out  = (float*)d_out;
  float* sums = out;                                   // [B][NV][5]
  float* cnts = out + kVoxElems;                       // [B][NV]

  // 1) zero the accumulation regions + write shape
  const long long nTotal = kVoxElems + kCntElems;      // divisible by 4
  voxel_zero_kernel<<<4096, 256, 0, stream>>>((float4*)out, nTotal / 4,
                                              out, nTotal);

  // 2) scatter-add points (double-buffered async LDS staging)
  const int total = kB * kN;                           // 1,200,000
  voxel_scatter_kernel<<<512, 256, 0, stream>>>(pts, sums, cnts, total);

  // 3) divide sums by max(count, 1) in place
  const long long nvoxTotal = kCntElems;
  voxel_finalize_kernel<<<(unsigned)((nvoxTotal + 255) / 256), 256, 0, stream>>>(
      sums, cnts, nvoxTotal);
}